// HardNegativeMiningLoss_78434692760098
// MI455X (gfx1250) — compile-verified
//
#include <hip/hip_runtime.h>
#include <stdint.h>

#define THREADS 256
#define WAVES   8              // wave32: 256/32
#define TILE    1024           // elements per tile (256 lanes x 4)
#define TILE_BYTES 4096
#define GRID    2048

// ---------------- global control block (zeroed every launch) ----------------
struct Ctl {
  uint32_t hist[4][256];   // per-level radix histograms
  uint32_t num_pos;
  uint32_t k;
  uint32_t k_rem;
  uint32_t cnt_gt;
  uint32_t prefix;         // left-aligned matched bits
  uint32_t pad_;
  double   pos_sum;        // sum of losses over positives
  double   sum_hi;         // sum of neg losses with top24 bits > prefix24
  double   sumBin[256];    // level-3 per-bin sums (top24 == prefix24)
};
__device__ Ctl g_ctl;

// CAS-loop double atomic (global memory; no f64 HW-atomic assumptions)
__device__ __forceinline__ void atomicAddD(double* addr, double val) {
  unsigned long long* a = (unsigned long long*)addr;
  unsigned long long old = *a, assumed;
  do {
    assumed = old;
    double cur = __longlong_as_double((long long)assumed);
    old = atomicCAS(a, assumed,
                    (unsigned long long)__double_as_longlong(cur + val));
  } while (old != assumed);
}

// numerically stable BCE with logits: max(x,0) - x*t + log1p(exp(-|x|))
__device__ __forceinline__ float bce_loss(float x, float t) {
  return fmaxf(x, 0.0f) - x * t + log1pf(expf(-fabsf(x)));
}

__global__ void hnm_init() {
  uint32_t* p = (uint32_t*)&g_ctl;
  const int total = (int)(sizeof(Ctl) / 4);
  for (int i = threadIdx.x; i < total; i += blockDim.x) p[i] = 0u;
}

// ============ pass A: read pred/tgt, histogram one level, optional bits out ==
template <int LEVEL>
__global__ __launch_bounds__(THREADS)
void hnm_pass(const float* __restrict__ pred, const float* __restrict__ tgt,
              int n, uint32_t* __restrict__ bits_out) {
  __shared__ float    sh_pred[2][TILE];
  __shared__ float    sh_tgt[2][TILE];
  __shared__ uint32_t sh_hist[256 * WAVES];   // bin-major, wave-padded
  __shared__ double   sh_red[THREADS];
  __shared__ uint32_t sh_posCnt;

  const int tid  = threadIdx.x;
  const int wave = tid >> 5;

  for (int i = tid; i < 256 * WAVES; i += THREADS) sh_hist[i] = 0u;
  if (tid == 0) sh_posCnt = 0u;
  __syncthreads();

  const uint32_t prefix = (LEVEL > 0) ? g_ctl.prefix : 0u;
  const uint32_t mask   = (LEVEL > 0) ? (0xFFFFFFFFu << (32 - 8 * LEVEL)) : 0u;
  const int      shift  = 24 - 8 * LEVEL;

  uint32_t posCnt = 0;
  double   posSum = 0.0;
  double   sumHi  = 0.0;

  auto process = [&](float x, float t) -> uint32_t {
    float loss = bce_loss(x, t);
    bool  pos  = t > 0.5f;
    if (LEVEL == 0) {
      if (pos) { posCnt++; posSum += (double)loss; return 0u; }
      uint32_t bits = __float_as_uint(loss);
      atomicAdd(&sh_hist[(bits >> 24) * WAVES + wave], 1u);
      return bits;
    } else {
      if (pos) return 0u;
      uint32_t bits = __float_as_uint(loss);
      uint32_t p = bits & mask;
      if (p == prefix) {
        uint32_t bin = (bits >> shift) & 0xFFu;
        atomicAdd(&sh_hist[bin * WAVES + wave], 1u);
        if (LEVEL == 3) atomicAddD(&g_ctl.sumBin[bin], (double)loss);
      } else if (LEVEL == 3 && p > prefix) {
        sumHi += (double)loss;
      }
      return bits;
    }
  };

  // ---- CDNA5 async-to-LDS double-buffered streaming (per-lane slots) ----
  const uint32_t numTiles = (uint32_t)(n / TILE);
  const uint64_t pbase  = (uint64_t)(uintptr_t)pred;
  const uint64_t tbase  = (uint64_t)(uintptr_t)tgt;
  const uint32_t lane16 = (uint32_t)tid * 16u;

  uint32_t ldsP[2], ldsT[2];
  ldsP[0] = (uint32_t)(uintptr_t)&sh_pred[0][0] + lane16;
  ldsP[1] = (uint32_t)(uintptr_t)&sh_pred[1][0] + lane16;
  ldsT[0] = (uint32_t)(uintptr_t)&sh_tgt[0][0] + lane16;
  ldsT[1] = (uint32_t)(uintptr_t)&sh_tgt[1][0] + lane16;

  auto issue = [&](int buf, uint32_t tile) {
    uint32_t off = tile * TILE_BYTES + lane16;
    asm volatile("global_load_async_to_lds_b128 %0, %1, %2"
                 :: "v"(ldsP[buf]), "v"(off), "s"(pbase) : "memory");
    asm volatile("global_load_async_to_lds_b128 %0, %1, %2"
                 :: "v"(ldsT[buf]), "v"(off), "s"(tbase) : "memory");
  };

  uint32_t t0  = blockIdx.x;
  int      buf = 0;
  if (t0 < numTiles) issue(0, t0);
  for (uint32_t ti = t0; ti < numTiles; ti += gridDim.x) {
    uint32_t nt = ti + gridDim.x;
    bool hasNext = nt < numTiles;
    if (hasNext) {
      issue(buf ^ 1, nt);
      asm volatile("s_wait_asynccnt 0x2" ::: "memory");  // current tile done
    } else {
      asm volatile("s_wait_asynccnt 0x0" ::: "memory");
    }
    float4 p4 = *(const float4*)&sh_pred[buf][tid * 4];
    float4 t4 = *(const float4*)&sh_tgt[buf][tid * 4];
    uint32_t bx = process(p4.x, t4.x);
    uint32_t by = process(p4.y, t4.y);
    uint32_t bz = process(p4.z, t4.z);
    uint32_t bw = process(p4.w, t4.w);
    if (LEVEL == 0 && bits_out) {
      uint4 v; v.x = bx; v.y = by; v.z = bz; v.w = bw;
      *(uint4*)&bits_out[ti * TILE + tid * 4] = v;
    }
    buf ^= 1;
  }

  // tail (n not a multiple of TILE): block 0, direct loads
  if (blockIdx.x == 0) {
    for (int i = (int)(numTiles * TILE) + tid; i < n; i += THREADS) {
      uint32_t bb = process(pred[i], tgt[i]);
      if (LEVEL == 0 && bits_out) bits_out[i] = bb;
    }
  }

  __syncthreads();

  // merge per-wave histograms -> global
  for (int b = tid; b < 256; b += THREADS) {
    uint32_t s = 0;
    for (int w = 0; w < WAVES; ++w) s += sh_hist[b * WAVES + w];
    if (s) atomicAdd(&g_ctl.hist[LEVEL][b], s);
  }

  if (LEVEL == 0) {
    if (posCnt) atomicAdd(&sh_posCnt, posCnt);
    sh_red[tid] = posSum;
    __syncthreads();
    for (int s = THREADS / 2; s > 0; s >>= 1) {
      if (tid < s) sh_red[tid] += sh_red[tid + s];
      __syncthreads();
    }
    if (tid == 0) {
      if (sh_posCnt) atomicAdd(&g_ctl.num_pos, sh_posCnt);
      if (sh_red[0] != 0.0) atomicAddD(&g_ctl.pos_sum, sh_red[0]);
    }
  }
  if (LEVEL == 3) {
    sh_red[tid] = sumHi;
    __syncthreads();
    for (int s = THREADS / 2; s > 0; s >>= 1) {
      if (tid < s) sh_red[tid] += sh_red[tid + s];
      __syncthreads();
    }
    if (tid == 0 && sh_red[0] != 0.0) atomicAddD(&g_ctl.sum_hi, sh_red[0]);
  }
}

// ===== pass B: stream stored loss-bits (half the traffic, no exp/log) =======
template <int LEVEL>
__global__ __launch_bounds__(THREADS)
void hnm_pass_bits(const uint32_t* __restrict__ bits, int n) {
  __shared__ uint32_t sh_bits[2][TILE];
  __shared__ uint32_t sh_hist[256 * WAVES];
  __shared__ double   sh_red[THREADS];

  const int tid  = threadIdx.x;
  const int wave = tid >> 5;

  for (int i = tid; i < 256 * WAVES; i += THREADS) sh_hist[i] = 0u;
  __syncthreads();

  const uint32_t prefix = g_ctl.prefix;
  const uint32_t mask   = 0xFFFFFFFFu << (32 - 8 * LEVEL);
  const int      shift  = 24 - 8 * LEVEL;

  double sumHi = 0.0;

  auto proc = [&](uint32_t b) {
    // positives were stored as 0; they can only pollute the all-zero bin,
    // whose threshold value is exactly 0.0 -> zero contribution (exact).
    uint32_t p = b & mask;
    if (p == prefix) {
      uint32_t bin = (b >> shift) & 0xFFu;
      atomicAdd(&sh_hist[bin * WAVES + wave], 1u);
      if (LEVEL == 3) atomicAddD(&g_ctl.sumBin[bin], (double)__uint_as_float(b));
    } else if (LEVEL == 3 && p > prefix) {
      sumHi += (double)__uint_as_float(b);
    }
  };

  const uint32_t numTiles = (uint32_t)(n / TILE);
  const uint64_t base   = (uint64_t)(uintptr_t)bits;
  const uint32_t lane16 = (uint32_t)tid * 16u;

  uint32_t ldsB[2];
  ldsB[0] = (uint32_t)(uintptr_t)&sh_bits[0][0] + lane16;
  ldsB[1] = (uint32_t)(uintptr_t)&sh_bits[1][0] + lane16;

  auto issue = [&](int buf, uint32_t tile) {
    uint32_t off = tile * TILE_BYTES + lane16;
    asm volatile("global_load_async_to_lds_b128 %0, %1, %2"
                 :: "v"(ldsB[buf]), "v"(off), "s"(base) : "memory");
  };

  uint32_t t0  = blockIdx.x;
  int      buf = 0;
  if (t0 < numTiles) issue(0, t0);
  for (uint32_t ti = t0; ti < numTiles; ti += gridDim.x) {
    uint32_t nt = ti + gridDim.x;
    bool hasNext = nt < numTiles;
    if (hasNext) {
      issue(buf ^ 1, nt);
      asm volatile("s_wait_asynccnt 0x1" ::: "memory");
    } else {
      asm volatile("s_wait_asynccnt 0x0" ::: "memory");
    }
    uint4 b4 = *(const uint4*)&sh_bits[buf][tid * 4];
    proc(b4.x); proc(b4.y); proc(b4.z); proc(b4.w);
    buf ^= 1;
  }

  if (blockIdx.x == 0) {
    for (int i = (int)(numTiles * TILE) + tid; i < n; i += THREADS)
      proc(bits[i]);
  }

  __syncthreads();
  for (int b = tid; b < 256; b += THREADS) {
    uint32_t s = 0;
    for (int w = 0; w < WAVES; ++w) s += sh_hist[b * WAVES + w];
    if (s) atomicAdd(&g_ctl.hist[LEVEL][b], s);
  }
  if (LEVEL == 3) {
    sh_red[tid] = sumHi;
    __syncthreads();
    for (int s = THREADS / 2; s > 0; s >>= 1) {
      if (tid < s) sh_red[tid] += sh_red[tid + s];
      __syncthreads();
    }
    if (tid == 0 && sh_red[0] != 0.0) atomicAddD(&g_ctl.sum_hi, sh_red[0]);
  }
}

// ---------------- tiny per-level selection / finalize ----------------
__global__ void hnm_select(int level, int n, float* out) {
  if (threadIdx.x != 0 || blockIdx.x != 0) return;
  if (level == 0) {
    uint32_t np = g_ctl.num_pos;
    uint32_t nn = (uint32_t)n - np;
    int kf = (int)((float)np * 3.0f);          // matches jnp f32 mult + i32 cast
    uint32_t k = (uint32_t)(kf > 100 ? kf : 100);
    if (k > nn) k = nn;
    g_ctl.k = k; g_ctl.k_rem = k; g_ctl.cnt_gt = 0;
    g_ctl.prefix = (k == 0) ? 0xFFFFFFFFu : 0u;   // match-nothing sentinel
  }
  uint32_t k = g_ctl.k;
  if (k == 0) {
    if (level == 3) {
      uint32_t cnt = g_ctl.num_pos;
      *out = (cnt > 0) ? (float)(g_ctl.pos_sum / (double)cnt) : 0.0f;
    }
    return;
  }
  uint32_t k_rem = g_ctl.k_rem;
  uint32_t cum = 0;
  int b = 255;
  for (; b > 0; --b) {
    uint32_t h = g_ctl.hist[level][b];
    if (cum + h >= k_rem) break;
    cum += h;
  }
  g_ctl.cnt_gt += cum;
  k_rem -= cum;
  g_ctl.k_rem = k_rem;
  g_ctl.prefix |= ((uint32_t)b) << (24 - 8 * level);
  if (level == 3) {
    double gt = g_ctl.sum_hi;
    for (int i = b + 1; i < 256; ++i) gt += g_ctl.sumBin[i];
    float  T   = __uint_as_float(g_ctl.prefix);          // exact k-th value
    double sel = g_ctl.pos_sum + gt + (double)k_rem * (double)T;
    uint32_t cnt = g_ctl.num_pos + k;
    *out = (cnt > 0) ? (float)(sel / (double)cnt) : 0.0f;
  }
}

extern "C" void kernel_launch(void* const* d_in, const int* in_sizes, int n_in,
                              void* d_out, int out_size, void* d_ws, size_t ws_size,
                              hipStream_t stream) {
  (void)n_in; (void)out_size;
  const float* pred = (const float*)d_in[0];
  const float* tgt  = (const float*)d_in[1];
  float* out = (float*)d_out;
  const int n = in_sizes[0];

  const bool useBits = (d_ws != nullptr) && (ws_size >= (size_t)n * 4u);
  uint32_t* bitsArr = useBits ? (uint32_t*)d_ws : nullptr;

  hipLaunchKernelGGL(hnm_init, dim3(1), dim3(256), 0, stream);

  hipLaunchKernelGGL(hnm_pass<0>, dim3(GRID), dim3(THREADS), 0, stream,
                     pred, tgt, n, bitsArr);
  hipLaunchKernelGGL(hnm_select, dim3(1), dim3(1), 0, stream, 0, n, out);

  if (useBits) {
    hipLaunchKernelGGL(hnm_pass_bits<1>, dim3(GRID), dim3(THREADS), 0, stream, bitsArr, n);
    hipLaunchKernelGGL(hnm_select, dim3(1), dim3(1), 0, stream, 1, n, out);
    hipLaunchKernelGGL(hnm_pass_bits<2>, dim3(GRID), dim3(THREADS), 0, stream, bitsArr, n);
    hipLaunchKernelGGL(hnm_select, dim3(1), dim3(1), 0, stream, 2, n, out);
    hipLaunchKernelGGL(hnm_pass_bits<3>, dim3(GRID), dim3(THREADS), 0, stream, bitsArr, n);
    hipLaunchKernelGGL(hnm_select, dim3(1), dim3(1), 0, stream, 3, n, out);
  } else {
    hipLaunchKernelGGL(hnm_pass<1>, dim3(GRID), dim3(THREADS), 0, stream,
                       pred, tgt, n, (uint32_t*)nullptr);
    hipLaunchKernelGGL(hnm_select, dim3(1), dim3(1), 0, stream, 1, n, out);
    hipLaunchKernelGGL(hnm_pass<2>, dim3(GRID), dim3(THREADS), 0, stream,
                       pred, tgt, n, (uint32_t*)nullptr);
    hipLaunchKernelGGL(hnm_select, dim3(1), dim3(1), 0, stream, 2, n, out);
    hipLaunchKernelGGL(hnm_pass<3>, dim3(GRID), dim3(THREADS), 0, stream,
                       pred, tgt, n, (uint32_t*)nullptr);
    hipLaunchKernelGGL(hnm_select, dim3(1), dim3(1), 0, stream, 3, n, out);
  }
}